// SparseFeedForward_47425028882858
// MI455X (gfx1250) — compile-verified
//
#include <hip/hip_runtime.h>
#include <hip/hip_bf16.h>
#include <math.h>

// ---------------------------------------------------------------------------
// SparseFeedForward on MI455X (gfx1250, wave32).
//   DIM=4096, INTER=14336, SPARSITY=256 (56 blocks), RANK=16, BT=256.
// Dominant cost: 469 MB of gathered w1/w2t rows (~20us at 23.3 TB/s HBM).
// Controller GEMMs are tiny -> run them on the WMMA pipe in f32 so the
// argmax neuron selection keeps full precision.
// ---------------------------------------------------------------------------

typedef float v2f __attribute__((ext_vector_type(2)));
typedef float v8f __attribute__((ext_vector_type(8)));

#define DIM    4096
#define INTER  14336
#define NBLK   56
#define SPARS  256
#define RANK   16
#define BT     256

// ---------------------------------------------------------------------------
// Phase 1: r[256,16] = x[256,4096] @ c1_w^T[4096,16]   (V_WMMA_F32_16X16X4_F32)
// One wave per 16-row M-tile. K loop 4096 in chunks of 4, 4 independent
// accumulators to break the WMMA D->C dependency chain.
//
// A 16x4 f32 layout (ISA 7.12.2): lanes 0-15 hold M=lane, VGPR0/1 = K0/K1;
// lanes 16-31 hold M=lane-16, VGPR0/1 = K2/K3. B 4x16 assumed symmetric
// (N = lane&15, K = 2*(lane>>4) + vgpr).
// ---------------------------------------------------------------------------
__global__ __launch_bounds__(32) void ctrl_rank_kernel(
    const float* __restrict__ x, const float* __restrict__ c1w,
    float* __restrict__ r) {
  const int bt   = blockIdx.x;        // 0..15, M tile
  const int lane = threadIdx.x;       // 0..31
  const int half = lane >> 4;         // 0/1
  const int mr   = lane & 15;
  const int koff = half * 2;

  const float* xrow = x   + (size_t)(bt * 16 + mr) * DIM + koff; // A: M=mr
  const float* brow = c1w + (size_t)mr * DIM + koff;             // B: N=mr (rank row)

  v8f c0 = {}, c1 = {}, c2 = {}, c3 = {};
  for (int k = 0; k < DIM; k += 16) {
    v2f a0 = *(const v2f*)(xrow + k     );
    v2f b0 = *(const v2f*)(brow + k     );
    v2f a1 = *(const v2f*)(xrow + k +  4);
    v2f b1 = *(const v2f*)(brow + k +  4);
    v2f a2 = *(const v2f*)(xrow + k +  8);
    v2f b2 = *(const v2f*)(brow + k +  8);
    v2f a3 = *(const v2f*)(xrow + k + 12);
    v2f b3 = *(const v2f*)(brow + k + 12);
    c0 = __builtin_amdgcn_wmma_f32_16x16x4_f32(false, a0, false, b0, (short)0, c0, false, false);
    c1 = __builtin_amdgcn_wmma_f32_16x16x4_f32(false, a1, false, b1, (short)0, c1, false, false);
    c2 = __builtin_amdgcn_wmma_f32_16x16x4_f32(false, a2, false, b2, (short)0, c2, false, false);
    c3 = __builtin_amdgcn_wmma_f32_16x16x4_f32(false, a3, false, b3, (short)0, c3, false, false);
  }
  v8f c = c0 + c1 + c2 + c3;
  // D layout: M = v + 8*half, N = lane&15
#pragma unroll
  for (int v = 0; v < 8; ++v)
    r[(size_t)(bt * 16 + v + 8 * half) * RANK + mr] = c[v];
}

// ---------------------------------------------------------------------------
// Phase 2: logits tile + blockwise argmax.
// Grid: (56 blocks, 16 b-tiles), one wave each. For each of 16 N-tiles:
// 4 chained x4-WMMAs cover K=16; D staged to LDS as [16 rows][256 neurons];
// then lanes 0-15 argmax their row -> absolute neuron id.
// ---------------------------------------------------------------------------
__global__ __launch_bounds__(32) void ctrl_select_kernel(
    const float* __restrict__ r, const float* __restrict__ c2w,
    int* __restrict__ idx) {
  const int blk  = blockIdx.x;        // 0..55
  const int bt   = blockIdx.y;        // 0..15
  const int lane = threadIdx.x;
  const int half = lane >> 4;
  const int nr   = lane & 15;
  const int koff = half * 2;

  __shared__ float tile[16 * SPARS];  // 16 KB

  // A operand (r tile, [16 b][16 k]) reused for all N-tiles.
  v2f a[4];
#pragma unroll
  for (int kc = 0; kc < 4; ++kc)
    a[kc] = *(const v2f*)(r + (size_t)(bt * 16 + nr) * RANK + 4 * kc + koff);

  for (int nt = 0; nt < 16; ++nt) {
    const int n0 = blk * SPARS + nt * 16 + nr;                 // B column = neuron
    const float* brow = c2w + (size_t)n0 * RANK + koff;        // B[k][n] = c2w[n][k]
    v8f c = {};
#pragma unroll
    for (int kc = 0; kc < 4; ++kc) {
      v2f bv = *(const v2f*)(brow + 4 * kc);
      c = __builtin_amdgcn_wmma_f32_16x16x4_f32(false, a[kc], false, bv, (short)0, c, false, false);
    }
#pragma unroll
    for (int v = 0; v < 8; ++v)
      tile[(v + 8 * half) * SPARS + nt * 16 + nr] = c[v];
  }
  __syncthreads();  // drain DScnt before cross-lane readback

  if (lane < 16) {
    const float* row = tile + lane * SPARS;
    float best = -INFINITY;
    int bi = 0;
    for (int n = 0; n < SPARS; ++n) {
      float v = row[n];
      if (v > best) { best = v; bi = n; }   // strict '>' keeps first max (jnp.argmax)
    }
    idx[(size_t)(bt * 16 + lane) * NBLK + blk] = blk * SPARS + bi;
  }
}

// ---------------------------------------------------------------------------
// Phase 3: bandwidth-bound gather/dot/scatter. One WG (1024 thr = 32 waves)
// per token row b; each thread owns one float4 slice of the 4096-wide row.
// Two neurons per iteration: all four 16KB gathered rows (w1/w2t x 2) are
// issued before the reduction so 64B/thread is in flight across the barrier
// stall, and the shuffle tree reduces both partials in one pass (halves the
// barrier count). Next iteration's rows are prefetched (global_prefetch_b8).
// ---------------------------------------------------------------------------
__global__ __launch_bounds__(1024) void sparse_mlp_kernel(
    const float* __restrict__ x, const float* __restrict__ w1,
    const float* __restrict__ w2t, const int* __restrict__ idx,
    float* __restrict__ out) {
  const int b    = blockIdx.x;
  const int tid  = threadIdx.x;       // 0..1023
  const int lane = tid & 31;
  const int wave = tid >> 5;          // 0..31

  __shared__ float red0[32];
  __shared__ float red1[32];
  __shared__ float hbc0, hbc1;
  __shared__ int   sidx[NBLK];

  if (tid < NBLK) sidx[tid] = idx[(size_t)b * NBLK + tid];
  __syncthreads();

  const int t4 = tid * 4;
  const float4 xv = *(const float4*)(x + (size_t)b * DIM + t4);
  float4 acc = make_float4(0.f, 0.f, 0.f, 0.f);

  for (int k = 0; k < NBLK; k += 2) {
    const size_t ra = (size_t)sidx[k] * DIM;
    const size_t rb = (size_t)sidx[k + 1] * DIM;
    // Issue all four gathered row loads up front (max MLP before the stall).
    const float4 w1a = *(const float4*)(w1  + ra + t4);
    const float4 w1b = *(const float4*)(w1  + rb + t4);
    const float4 w2a = *(const float4*)(w2t + ra + t4);
    const float4 w2b = *(const float4*)(w2t + rb + t4);
    if (k + 2 < NBLK) {
      __builtin_prefetch(w1  + (size_t)sidx[k + 2] * DIM + t4, 0, 1);
      __builtin_prefetch(w2t + (size_t)sidx[k + 2] * DIM + t4, 0, 1);
      __builtin_prefetch(w1  + (size_t)sidx[k + 3] * DIM + t4, 0, 1);
      __builtin_prefetch(w2t + (size_t)sidx[k + 3] * DIM + t4, 0, 1);
    }

    float p0 = w1a.x * xv.x + w1a.y * xv.y + w1a.z * xv.z + w1a.w * xv.w;
    float p1 = w1b.x * xv.x + w1b.y * xv.y + w1b.z * xv.z + w1b.w * xv.w;
#pragma unroll
    for (int off = 16; off > 0; off >>= 1) {
      p0 += __shfl_down(p0, off, 32);
      p1 += __shfl_down(p1, off, 32);
    }
    if (lane == 0) { red0[wave] = p0; red1[wave] = p1; }
    __syncthreads();
    if (wave == 0) {
      float v0 = red0[lane];
      float v1 = red1[lane];
#pragma unroll
      for (int off = 16; off > 0; off >>= 1) {
        v0 += __shfl_down(v0, off, 32);
        v1 += __shfl_down(v1, off, 32);
      }
      if (lane == 0) {
        hbc0 = v0 > 0.f ? v0 : 0.f;   // ReLU
        hbc1 = v1 > 0.f ? v1 : 0.f;
      }
    }
    __syncthreads();
    const float h0 = hbc0;
    const float h1 = hbc1;

    acc.x += h0 * w2a.x + h1 * w2b.x;
    acc.y += h0 * w2a.y + h1 * w2b.y;
    acc.z += h0 * w2a.z + h1 * w2b.z;
    acc.w += h0 * w2a.w + h1 * w2b.w;
  }
  *(float4*)(out + (size_t)b * DIM + t4) = acc;
}

// ---------------------------------------------------------------------------
// Launch. Inputs: x, c1_w, c2_w, w1, w2t (all f32). Output: [256,1,4096] f32.
// Workspace: r (256*16 f32) then idx (256*56 i32) -> ~74 KB.
// ---------------------------------------------------------------------------
extern "C" void kernel_launch(void* const* d_in, const int* in_sizes, int n_in,
                              void* d_out, int out_size, void* d_ws, size_t ws_size,
                              hipStream_t stream) {
  const float* x   = (const float*)d_in[0];
  const float* c1w = (const float*)d_in[1];
  const float* c2w = (const float*)d_in[2];
  const float* w1  = (const float*)d_in[3];
  const float* w2t = (const float*)d_in[4];
  float* out = (float*)d_out;

  float* r   = (float*)d_ws;                                   // 16 KB
  int*   idx = (int*)((char*)d_ws + (size_t)BT * RANK * sizeof(float));

  ctrl_rank_kernel<<<dim3(BT / 16), dim3(32), 0, stream>>>(x, c1w, r);
  ctrl_select_kernel<<<dim3(NBLK, BT / 16), dim3(32), 0, stream>>>(r, c2w, idx);
  sparse_mlp_kernel<<<dim3(BT), dim3(1024), 0, stream>>>(x, w1, w2t, idx, out);
}